// MaskedAttention_64785286693220
// MI455X (gfx1250) — compile-verified
//
#include <hip/hip_runtime.h>

typedef __attribute__((ext_vector_type(2))) float v2f;
typedef __attribute__((ext_vector_type(8))) float v8f;

#define E_   512
#define H_   8
#define D_   64
#define WIN_ 128
#define L_   4096
#define B_   2
#define R_   (B_ * L_)
#define F3_  (3 * E_)

// V_WMMA_F32_16X16X4_F32 : D(16x16 f32) = A(16x4 f32, 2 VGPR) * B(4x16 f32, 2 VGPR) + C
// A: lane = M + (K>=2 ? 16 : 0), vgpr = K&1 ; B mirrors with lane = N
// C/D: vgpr j, lanes0-15 -> M=j, lanes16-31 -> M=j+8, N = lane&15
__device__ __forceinline__ v8f wmma4(v2f a, v2f b, v8f c) {
    return __builtin_amdgcn_wmma_f32_16x16x4_f32(false, a, false, b, (short)0, c,
                                                 false, false);
}

// C[R,F] = A[R,512] * W[F,512]^T + bias[F].
// One wave computes a 64x16 tile: 4 row-tiles share each B fragment
// (4 independent WMMA chains), with software-pipelined fragment loads.
__global__ void __launch_bounds__(128)
gemm_bias_nt(const float* __restrict__ A, const float* __restrict__ W,
             const float* __restrict__ bias, float* __restrict__ C,
             int rTiles, int F) {
    const int wave = threadIdx.x >> 5;
    const int lane = threadIdx.x & 31;
    const int half = lane >> 4;
    const int l16  = lane & 15;

    const int cTiles = F >> 4;
    const int tile   = blockIdx.x * 4 + wave;
    if (tile >= rTiles * cTiles) return;
    const int rt = tile / cTiles;
    const int ct = tile % cTiles;
    const int r0 = rt << 6, c0 = ct << 4;

    const float* a0 = A + (size_t)(r0 + l16) * E_ + 2 * half;
    const float* a1 = a0 + (size_t)16 * E_;
    const float* a2 = a0 + (size_t)32 * E_;
    const float* a3 = a0 + (size_t)48 * E_;
    const float* br = W + (size_t)(c0 + l16) * E_ + 2 * half;

    const v8f vz = {};
    v8f acc0 = vz, acc1 = vz, acc2 = vz, acc3 = vz;

    // software pipeline: cur fragments compute while nxt fragments load
    v2f bc  = *(const v2f*)br;
    v2f ac0 = *(const v2f*)a0;
    v2f ac1 = *(const v2f*)a1;
    v2f ac2 = *(const v2f*)a2;
    v2f ac3 = *(const v2f*)a3;

#pragma unroll 2
    for (int k = 0; k < E_ - 4; k += 4) {
        const int kn = k + 4;
        v2f bn  = *(const v2f*)(br + kn);
        v2f an0 = *(const v2f*)(a0 + kn);
        v2f an1 = *(const v2f*)(a1 + kn);
        v2f an2 = *(const v2f*)(a2 + kn);
        v2f an3 = *(const v2f*)(a3 + kn);
        acc0 = wmma4(ac0, bc, acc0);
        acc1 = wmma4(ac1, bc, acc1);
        acc2 = wmma4(ac2, bc, acc2);
        acc3 = wmma4(ac3, bc, acc3);
        bc = bn; ac0 = an0; ac1 = an1; ac2 = an2; ac3 = an3;
    }
    acc0 = wmma4(ac0, bc, acc0);
    acc1 = wmma4(ac1, bc, acc1);
    acc2 = wmma4(ac2, bc, acc2);
    acc3 = wmma4(ac3, bc, acc3);

    const float bcol = bias[c0 + l16];
    v8f accs[4] = {acc0, acc1, acc2, acc3};
#pragma unroll
    for (int i = 0; i < 4; ++i) {
        float* crow = C + (size_t)(r0 + 16 * i + 8 * half) * F + c0 + l16;
#pragma unroll
        for (int j = 0; j < 8; ++j)
            crow[(size_t)j * F] = accs[i][j] + bcol;
    }
}

// Banded flash attention over qkv[R,3E]; one wave = one (b,h,16-query) tile.
__global__ void __launch_bounds__(128)
attn_kernel(const float* __restrict__ qkv, float* __restrict__ O) {
    __shared__ float plds[4][16][17];  // per-wave P transpose staging (padded)

    const int wave = threadIdx.x >> 5;
    const int lane = threadIdx.x & 31;
    const int half = lane >> 4;
    const int l16  = lane & 15;

    const int QT   = L_ / 16;
    const int tile = blockIdx.x * 4 + wave;
    if (tile >= B_ * H_ * QT) return;
    const int qt = tile % QT;
    const int bh = tile / QT;
    const int b  = bh / H_;
    const int h  = bh % H_;
    const int q0 = qt * 16;

    // Preload Q A-fragments: aq[c].{x,y} = Q[q0+l16, 4c + 2*half + {0,1}]
    const float* qrow =
        qkv + (size_t)(b * L_ + q0 + l16) * F3_ + h * D_ + 2 * half;
    v2f aq[16];
#pragma unroll
    for (int c = 0; c < 16; ++c) aq[c] = *(const v2f*)(qrow + 4 * c);

    const v8f vz = {};
    v8f acc[4];
#pragma unroll
    for (int t = 0; t < 4; ++t) acc[t] = vz;
    float mst[8], lst[8];
#pragma unroll
    for (int j = 0; j < 8; ++j) { mst[j] = -__builtin_inff(); lst[j] = 0.f; }

    int ktlo = qt - 8;      if (ktlo < 0) ktlo = 0;
    int kthi = qt + 8;      if (kthi > QT - 1) kthi = QT - 1;

    float (*P)[17] = plds[wave];

    for (int kt = ktlo; kt <= kthi; ++kt) {
        const int k0 = kt * 16;

        const float* krow =
            qkv + (size_t)(b * L_ + k0 + l16) * F3_ + E_ + h * D_ + 2 * half;
        const float* vbase =
            qkv + (size_t)(b * L_ + k0) * F3_ + 2 * E_ + h * D_;

        // prefetch next k-tile's K row and V row (L2 hint)
        if (kt < kthi) {
            __builtin_prefetch(krow + (size_t)16 * F3_, 0, 0);
            __builtin_prefetch(vbase + (size_t)(16 + l16) * F3_, 0, 0);
        }

        // S = Q * K^T : batch-load all 16 B-fragments, two WMMA chains
        v2f bk[16];
#pragma unroll
        for (int c = 0; c < 16; ++c) bk[c] = *(const v2f*)(krow + 4 * c);
        v8f s0 = vz, s1 = vz;
#pragma unroll
        for (int c = 0; c < 8; ++c) {
            s0 = wmma4(aq[2 * c],     bk[2 * c],     s0);
            s1 = wmma4(aq[2 * c + 1], bk[2 * c + 1], s1);
        }
        v8f s = s0 + s1;

        // scale 1/sqrt(D) + band mask: blocked iff dist==0 || dist>WIN
        const int ki = k0 + l16;
#pragma unroll
        for (int j = 0; j < 8; ++j) {
            const int qi = q0 + j + 8 * half;
            int dist = qi - ki; if (dist < 0) dist = -dist;
            float sv = s[j] * 0.125f;
            if (dist == 0 || dist > WIN_) sv = -__builtin_inff();
            s[j] = sv;
        }

        // online softmax (row stats replicated across the 16 lanes of a half)
        float alpha[8];
        v8f p;
#pragma unroll
        for (int j = 0; j < 8; ++j) {
            float v = s[j];
            v = fmaxf(v, __shfl_xor(v, 1, 32));
            v = fmaxf(v, __shfl_xor(v, 2, 32));
            v = fmaxf(v, __shfl_xor(v, 4, 32));
            v = fmaxf(v, __shfl_xor(v, 8, 32));
            const float mnew = fmaxf(mst[j], v);
            const float al   = __expf(mst[j] - mnew);
            alpha[j] = al;
            mst[j]   = mnew;
            const float pv = __expf(s[j] - mnew);
            p[j] = pv;
            float rs = pv;
            rs += __shfl_xor(rs, 1, 32);
            rs += __shfl_xor(rs, 2, 32);
            rs += __shfl_xor(rs, 4, 32);
            rs += __shfl_xor(rs, 8, 32);
            lst[j] = lst[j] * al + rs;
        }

        // rescale running accumulators
#pragma unroll
        for (int t = 0; t < 4; ++t)
#pragma unroll
            for (int j = 0; j < 8; ++j) acc[t][j] *= alpha[j];

        // C-layout -> A-layout transpose of P through LDS (wave-private tile)
#pragma unroll
        for (int j = 0; j < 8; ++j) P[j + 8 * half][l16] = p[j];

        // Batch-load all V B-fragments + P A-fragments, then 4 chains (t)
        v2f bv[4][4];  // [c][t]
#pragma unroll
        for (int c = 0; c < 4; ++c) {
            const float* vp = vbase + (size_t)(4 * c + 2 * half) * F3_ + l16;
#pragma unroll
            for (int t = 0; t < 4; ++t) {
                bv[c][t].x = vp[t * 16];
                bv[c][t].y = vp[t * 16 + F3_];
            }
        }
        v2f ap[4];
#pragma unroll
        for (int c = 0; c < 4; ++c) {
            ap[c].x = P[l16][4 * c + 2 * half];
            ap[c].y = P[l16][4 * c + 2 * half + 1];
        }
#pragma unroll
        for (int c = 0; c < 4; ++c)
#pragma unroll
            for (int t = 0; t < 4; ++t)
                acc[t] = wmma4(ap[c], bv[c][t], acc[t]);
    }

    // normalize and write O[R,E] in (l, h*D+d) layout for the final GEMM
    float* orow = O + (size_t)(b * L_ + q0 + 8 * half) * E_ + h * D_;
#pragma unroll
    for (int j = 0; j < 8; ++j) {
        const float inv = 1.0f / lst[j];
#pragma unroll
        for (int t = 0; t < 4; ++t)
            orow[(size_t)j * E_ + t * 16 + l16] = acc[t][j] * inv;
    }
}

extern "C" void kernel_launch(void* const* d_in, const int* in_sizes, int n_in,
                              void* d_out, int out_size, void* d_ws, size_t ws_size,
                              hipStream_t stream) {
    const float* x  = (const float*)d_in[0];  // [B,L,E]
    const float* w1 = (const float*)d_in[1];  // [3E,E]
    const float* b1 = (const float*)d_in[2];  // [3E]
    const float* w2 = (const float*)d_in[3];  // [E,E]
    const float* b2 = (const float*)d_in[4];  // [E]
    float* out = (float*)d_out;               // [B,L,E]

    float* qkv  = (float*)d_ws;               // R x 3E  (48 MB)
    float* Obuf = qkv + (size_t)R_ * F3_;     // R x E   (16 MB)

    {
        const int tiles = (R_ / 64) * (F3_ / 16);   // 128 * 96
        gemm_bias_nt<<<dim3(tiles / 4), dim3(128), 0, stream>>>(
            x, w1, b1, qkv, R_ / 64, F3_);
    }
    {
        const int tiles = B_ * H_ * (L_ / 16);      // 4096
        attn_kernel<<<dim3(tiles / 4), dim3(128), 0, stream>>>(qkv, Obuf);
    }
    {
        const int tiles = (R_ / 64) * (E_ / 16);    // 128 * 32
        gemm_bias_nt<<<dim3(tiles / 4), dim3(128), 0, stream>>>(
            Obuf, w2, b2, out, R_ / 64, E_);
    }
}